// MultiInterestExtractor_3435973837412
// MI455X (gfx1250) — compile-verified
//
#include <hip/hip_runtime.h>
#include <hip/hip_bf16.h>
#include <math.h>

// MultiInterestExtractor fused kernel for MI455X (gfx1250, wave32).
//   Phase 1 (per wave, 32 rows = 2 WMMA M-tiles):
//     s = x @ C^T via v_wmma_f32_16x16x4_f32 (fp32, K=64 in 16 steps)
//     z2 = (s + gumbel)/tau + 64  ->  padded LDS (row-major, stride 68)
//   Phase 2 (one row per lane, no cross-lane ops):
//     softmax needs no max-sub (tau=10 bounds |z|); sum of exp in-register.
//     top-10 via u32-packed keys (value bits | 63-idx) + branch-free
//     sorted-insertion network (v_max_u32/v_min_u32 only).
// Traffic: ~486 MB -> ~21 us floor at 23.3 TB/s; VALU/DS now sized under that.

typedef float v2f __attribute__((ext_vector_type(2)));
typedef float v4f __attribute__((ext_vector_type(4)));
typedef float v8f __attribute__((ext_vector_type(8)));

#define HIDDEN   64
#define ASPECTS  64
#define NROWS    (4096 * 200)     // BATCH * SEQ
#define TOPK     10
#define LDS_STRIDE 68             // 64 + 4 pad: 16B-aligned rows, spreads banks
#define ROWS_PER_WAVE 32
#define WAVES_PER_BLOCK 8

__global__ __launch_bounds__(256) void mie_topk_kernel(
    const float* __restrict__ x,    // [NROWS, HIDDEN]
    const float* __restrict__ Cw,   // [ASPECTS, HIDDEN]
    const float* __restrict__ g,    // [NROWS, ASPECTS]
    float* __restrict__ out)        // [NROWS*TOPK weights | NROWS*TOPK indices]
{
    __shared__ float zbuf[WAVES_PER_BLOCK * ROWS_PER_WAVE * LDS_STRIDE];  // 68 KB

    const int  lane = (int)(threadIdx.x & 31);
    const int  wib  = (int)(threadIdx.x >> 5);
    const int  wave = (int)blockIdx.x * WAVES_PER_BLOCK + wib;
    const int  lm   = lane & 15;
    const int  lh   = lane >> 4;
    const long rowbase = (long)wave * ROWS_PER_WAVE;
    float* zw = &zbuf[wib * ROWS_PER_WAVE * LDS_STRIDE];

    const float invtau = 0.1f;

    // ---------------- phase 1: WMMA GEMM + bias, z2 -> LDS ----------------
    // 16x4 F32 A/B layout: lane holds M(or N)=lm, VGPR0/1 = K 2*lh / 2*lh+1,
    // so one float2 load at column (4*kk + 2*lh) fills a fragment.
    const float* brow = Cw + lm * HIDDEN + 2 * lh;   // B[k][n] = C[n][k]
#pragma unroll
    for (int t = 0; t < 2; ++t) {
        const float* xrow = x + (rowbase + t * 16 + lm) * HIDDEN + 2 * lh;
        v2f a[16];
#pragma unroll
        for (int kk = 0; kk < 16; ++kk)
            a[kk] = *(const v2f*)(xrow + 4 * kk);

        v8f acc[4] = {};
#pragma unroll
        for (int nt = 0; nt < 4; ++nt) {
            const float* bnt = brow + nt * 16 * HIDDEN;
#pragma unroll
            for (int kk = 0; kk < 16; ++kk) {
                v2f b = *(const v2f*)(bnt + 4 * kk);
                acc[nt] = __builtin_amdgcn_wmma_f32_16x16x4_f32(
                    false, a[kk], false, b, (short)0, acc[nt], false, false);
            }
        }

        // C/D layout: row = v + 8*lh, col = nt*16 + lm.  z2 = (s+g)/tau + 64
        // (+64 keeps z2 positive so u32 bit compare == float compare).
#pragma unroll
        for (int v = 0; v < 8; ++v) {
            const int rl = t * 16 + v + 8 * lh;
            const float* grow = g + (rowbase + rl) * ASPECTS + lm;
#pragma unroll
            for (int nt = 0; nt < 4; ++nt) {
                float z2 = fmaf(acc[nt][v] + grow[nt * 16], invtau, 64.0f);
                zw[rl * LDS_STRIDE + nt * 16 + lm] = z2;
            }
        }
    }

    __syncthreads();   // order LDS stores before cross-lane reads

    // ---------------- phase 2: one row per lane ----------------
    const float* zr = zw + lane * LDS_STRIDE;

    unsigned s[TOPK];
#pragma unroll
    for (int j = 0; j < TOPK; ++j) s[j] = 0u;

    float sum = 0.0f;
#pragma unroll
    for (int q = 0; q < 16; ++q) {
        v4f zq = *(const v4f*)(zr + q * 4);
#pragma unroll
        for (int c = 0; c < 4; ++c) {
            const int   i  = q * 4 + c;
            const float zv = zq[c];
            sum += __expf(zv - 64.0f);                 // exp(z), safe range
            // key: high bits = value ordering, low 6 bits = 63-idx
            // (ties break toward lower index, like lax.top_k)
            unsigned key = (__float_as_uint(zv) & 0xFFFFFFC0u) | (unsigned)(63 - i);
            // branch-free sorted insertion (descending): 2 VALU per slot
#pragma unroll
            for (int j = 0; j < TOPK; ++j) {
                unsigned hi = (key > s[j]) ? key : s[j];   // v_max_u32
                unsigned lo = (key > s[j]) ? s[j] : key;   // v_min_u32
                s[j] = hi;
                key  = lo;
            }
        }
    }

    const float inv = 1.0f / sum;
    const long  row = rowbase + lane;
    const long  idx_base = (long)NROWS * TOPK;
#pragma unroll
    for (int j = 0; j < TOPK; ++j) {
        const int   idx = 63 - (int)(s[j] & 63u);
        const float zx  = zr[idx];                    // exact value (indexed DS read)
        const float wj  = __expf(zx - 64.0f) * inv;   // exact fp32 softmax weight
        out[row * TOPK + j]            = wj;
        out[idx_base + row * TOPK + j] = (float)idx;
    }
}

extern "C" void kernel_launch(void* const* d_in, const int* in_sizes, int n_in,
                              void* d_out, int out_size, void* d_ws, size_t ws_size,
                              hipStream_t stream) {
    (void)in_sizes; (void)n_in; (void)out_size; (void)d_ws; (void)ws_size;
    const float* x  = (const float*)d_in[0];   // x_u        [4096,200,64] fp32
    const float* Cw = (const float*)d_in[1];   // C_weight   [64,64]       fp32
    const float* g  = (const float*)d_in[2];   // gumbel     [4096,200,64] fp32
    float* out = (float*)d_out;                // [R*10 weights | R*10 indices(as float)]

    const int waves  = NROWS / ROWS_PER_WAVE;        // 25600 (exact)
    dim3 grid(waves / WAVES_PER_BLOCK);              // 3200  (exact -> EXEC all-1s)
    dim3 block(256);
    mie_topk_kernel<<<grid, block, 0, stream>>>(x, Cw, g, out);
}